// HyperbolicPrefixContrastiveLoss_60748017434921
// MI455X (gfx1250) — compile-verified
//
#include <hip/hip_runtime.h>
#include <hip/hip_bf16.h>
#include <math.h>

typedef float v2f __attribute__((ext_vector_type(2)));
typedef float v8f __attribute__((ext_vector_type(8)));

#define NROW 4096
#define DIM  512
#define PDIM 64
#define NT   256                          // 16-row tiles per dim (projector)
#define NT2  128                          // 32-row blocks per dim (pair sweep)
#define TILES2 (NT2*(NT2+1)/2)            // 8256 upper-tri 32x32 blocks
#define WPB  8                            // waves per block in pair kernel
#define NBLK2 ((TILES2 + WPB - 1) / WPB)  // 1032
#define EPSF 1e-6f

// ---------------------------------------------------------------- WMMA helper
__device__ __forceinline__ v8f wmma_f32(v2f a, v2f b, v8f c) {
  // D = A(16x4) * B(4x16) + C(16x16), all f32, wave32
  return __builtin_amdgcn_wmma_f32_16x16x4_f32(false, a, false, b, (short)0, c,
                                               false, false);
}

// ---------------------------------------------------------------- init
__global__ void hp_init_kernel(unsigned* hist, unsigned* state) {
  int tid = threadIdx.x;
  for (int i = tid; i < 768; i += blockDim.x) hist[i] = 0u;
  if (tid < 32) state[tid] = 0u;
}

// ---------------------------------------------------------------- projector
// One block = 16 rows. 4 waves each own a 16-col tile of P=64.
// Stage1: H = relu(E@W1+b1) -> LDS. Stage2: X = H@W2+b2 -> LDS.
// Then ball projection, sq[], packed prefix codes[].
__global__ void hp_proj_kernel(const float* __restrict__ E,
                               const long long* __restrict__ cid,
                               const float* __restrict__ W1,
                               const float* __restrict__ b1,
                               const float* __restrict__ W2,
                               const float* __restrict__ b2,
                               float* __restrict__ Xo,
                               float* __restrict__ sqv,
                               unsigned* __restrict__ codes) {
  __shared__ float Hs[16 * 68];
  __shared__ float Xs[16 * 68];
  __shared__ float scl[16];
  const int tid  = threadIdx.x;
  const int lane = tid & 31;
  const int wid  = tid >> 5;           // 0..3
  const int m    = lane & 15;
  const int khi  = lane >> 4;          // 0 or 1
  const int rowBase = blockIdx.x * 16;
  const int colBase = wid * 16;

  // ---- stage 1: H tile [16 x 16] = E[16 x 512] @ W1[512 x 16]
  {
    v8f c = {};
    const float* ap = E  + (size_t)(rowBase + m) * DIM + 2 * khi;
    const float* bp = W1 + (size_t)(2 * khi) * PDIM + (colBase + m);
    #pragma unroll 4
    for (int k0 = 0; k0 < DIM; k0 += 4) {
      v2f a, b;
      a.x = ap[k0];
      a.y = ap[k0 + 1];
      b.x = bp[(size_t)k0 * PDIM];
      b.y = bp[(size_t)(k0 + 1) * PDIM];
      c = wmma_f32(a, b, c);
    }
    float bias = b1[colBase + m];
    #pragma unroll
    for (int r = 0; r < 8; ++r)
      Hs[(r + 8 * khi) * 68 + colBase + m] = fmaxf(c[r] + bias, 0.f);
  }
  __syncthreads();

  // ---- stage 2: X tile [16 x 16] = H[16 x 64] @ W2[64 x 16]
  {
    v8f c = {};
    #pragma unroll
    for (int k0 = 0; k0 < PDIM; k0 += 4) {
      int k = k0 + 2 * khi;
      v2f a, b;
      a.x = Hs[m * 68 + k];
      a.y = Hs[m * 68 + k + 1];
      b.x = W2[(size_t)k * PDIM + colBase + m];
      b.y = W2[(size_t)(k + 1) * PDIM + colBase + m];
      c = wmma_f32(a, b, c);
    }
    float bias = b2[colBase + m];
    #pragma unroll
    for (int r = 0; r < 8; ++r)
      Xs[(r + 8 * khi) * 68 + colBase + m] = c[r] + bias;
  }
  __syncthreads();

  // ---- row norms, ball projection scale, sq, prefix codes
  if (tid < 16) {
    float s = 0.f;
    #pragma unroll 8
    for (int n = 0; n < PDIM; ++n) {
      float v = Xs[tid * 68 + n];
      s += v * v;
    }
    float norm  = sqrtf(s + EPSF);
    float scale = 0.99f / fmaxf(norm, 0.99f);
    scl[tid] = scale;
    sqv[rowBase + tid] = s * scale * scale;
    const long long* cp = cid + (size_t)(rowBase + tid) * 4;
    unsigned code = (unsigned)(cp[0] & 7) | ((unsigned)(cp[1] & 7) << 3) |
                    ((unsigned)(cp[2] & 7) << 6) | ((unsigned)(cp[3] & 7) << 9);
    codes[rowBase + tid] = code;
  }
  __syncthreads();

  for (int idx = tid; idx < 16 * PDIM; idx += blockDim.x) {
    int mm = idx >> 6, nn = idx & 63;
    Xo[(size_t)(rowBase + mm) * PDIM + nn] = Xs[mm * 68 + nn] * scl[mm];
  }
}

// ---------------------------------------------------------------- block decode
__device__ __forceinline__ void hp_decode_tile(int t, int& ti, int& tj) {
  float ft  = (float)t;
  float s   = 2.f * NT2 + 1.f;
  int row   = (int)((s - sqrtf(s * s - 8.f * ft)) * 0.5f);
  if (row < 0) row = 0;
  if (row >= NT2) row = NT2 - 1;
  while (row > 0 && (row * NT2 - row * (row - 1) / 2) > t) --row;
  while (((row + 1) * NT2 - (row + 1) * row / 2) <= t) ++row;
  ti = row;
  tj = row + (t - (row * NT2 - row * (row - 1) / 2));
}

// ---------------------------------------------------------------- 32x32 gram
// 2x2 register blocking: 4 b64 loads feed 4 WMMAs per k-step (1:1),
// each A/B operand pair reused by two WMMAs.
__device__ __forceinline__ void hp_gram2x2(const float* __restrict__ X,
                                           int rowI, int rowJ, int lane,
                                           v8f c[2][2]) {
  const int m   = lane & 15;
  const int khi = lane >> 4;
  const float* a0 = X + (size_t)(rowI + m) * PDIM + 2 * khi;
  const float* a1 = a0 + (size_t)16 * PDIM;
  const float* b0 = X + (size_t)(rowJ + m) * PDIM + 2 * khi;
  const float* b1 = b0 + (size_t)16 * PDIM;
  #pragma unroll
  for (int k0 = 0; k0 < PDIM; k0 += 4) {
    v2f A0, A1, B0, B1;
    A0.x = a0[k0]; A0.y = a0[k0 + 1];
    A1.x = a1[k0]; A1.y = a1[k0 + 1];
    B0.x = b0[k0]; B0.y = b0[k0 + 1];
    B1.x = b1[k0]; B1.y = b1[k0 + 1];
    c[0][0] = wmma_f32(A0, B0, c[0][0]);
    c[0][1] = wmma_f32(A0, B1, c[0][1]);
    c[1][0] = wmma_f32(A1, B0, c[1][0]);
    c[1][1] = wmma_f32(A1, B1, c[1][1]);
  }
}

// ---------------------------------------------------------------- pair passes
// mode 0: radix histogram pass `pass` (bytes 3..0 of loss bit pattern)
// mode 1: final sum of values strictly above per-group threshold
__global__ void hp_pair_kernel(const float* __restrict__ X,
                               const float* __restrict__ sqv,
                               const unsigned* __restrict__ codes,
                               unsigned* __restrict__ hist,
                               unsigned* __restrict__ state,
                               float* __restrict__ partial,
                               int mode, int pass) {
  __shared__ unsigned shh[768];
  __shared__ float    wsum[WPB][3];
  __shared__ unsigned wcnt[WPB][3];
  const int tid  = threadIdx.x;
  const int lane = tid & 31;
  const int wid  = tid >> 5;
  if (mode == 0) {
    for (int i = tid; i < 768; i += blockDim.x) shh[i] = 0u;
  }
  __syncthreads();

  float    gs[3] = {0.f, 0.f, 0.f};
  unsigned gc[3] = {0u, 0u, 0u};
  const int t = blockIdx.x * WPB + wid;   // wave-uniform
  if (t < TILES2) {
    int ti, tj;
    hp_decode_tile(t, ti, tj);
    const int rowI = ti * 32, rowJ = tj * 32;
    __builtin_prefetch(codes + rowJ, 0, 3);
    __builtin_prefetch(sqv + rowJ, 0, 3);

    v8f g[2][2];
    g[0][0] = (v8f){}; g[0][1] = (v8f){}; g[1][0] = (v8f){}; g[1][1] = (v8f){};
    hp_gram2x2(X, rowI, rowJ, lane, g);

    const int m   = lane & 15;
    const int khi = lane >> 4;
    const int shift = 24 - 8 * pass;
    // uniform radix-select state, hoisted
    const unsigned pref0 = state[9],  pref1 = state[10], pref2 = state[11];
    const unsigned thr0  = state[12], thr1  = state[13], thr2  = state[14];
    const unsigned prefA[3] = {pref0, pref1, pref2};
    const unsigned thrA[3]  = {thr0, thr1, thr2};
    // (1-ratio_c)^1.5 for lcp = 0,1,2
    const float pw15[3] = {1.0f, 0.64951905284f, 0.35355339059f};

    #pragma unroll
    for (int bj = 0; bj < 2; ++bj) {
      const int j = rowJ + 16 * bj + m;
      const unsigned cj = codes[j];
      const float   sqj = sqv[j];
      #pragma unroll
      for (int bi = 0; bi < 2; ++bi) {
        const int ibase = rowI + 16 * bi + 8 * khi;
        #pragma unroll
        for (int r = 0; r < 8; ++r) {
          const int i = ibase + r;
          if (i < j) {
            const float sqi = sqv[i];
            const unsigned ci = codes[i];
            const unsigned x  = ci ^ cj;
            int lcp = (x & 0x007u) ? 0 :
                      (x & 0x038u) ? 1 :
                      (x & 0x1C0u) ? 2 :
                      (x & 0xE00u) ? 3 : 4;
            int   grp;
            float tgt, bw, ls;
            if (lcp == 4)      { grp = 0; tgt = 0.7f;  bw = 0.10f; ls = 4.f; }
            else if (lcp == 3) { grp = 1; tgt = 0.55f; bw = 0.08f; ls = 3.f; }
            else               { grp = 2; tgt = 0.4f + pw15[lcp] * 0.8f;
                                 bw = 0.25f; ls = 1.f; }
            float d2  = fmaxf(sqi + sqj - 2.f * g[bi][bj][r], 0.f);
            float den = fmaxf((1.f - sqi) * (1.f - sqj), EPSF);
            float arg = fmaxf(1.f + 2.f * d2 / den, 1.f + EPSF);
            float dist = logf(arg + sqrtf(fmaxf(arg * arg - 1.f, 0.f)));
            float a  = fmaxf(tgt - bw, 0.f);
            float bb = tgt + bw;
            float lo = fmaxf(a - dist, 0.f);
            float hi = fmaxf(dist - bb, 0.f);
            float loss = ls * lo * lo + hi * hi;
            unsigned bits = __float_as_uint(loss);
            if (mode == 0) {
              bool take = true;
              if (pass != 0) {
                int hs = 32 - 8 * pass;   // bits above current byte
                take = ((bits >> hs) == (prefA[grp] >> hs));
              }
              if (take)
                atomicAdd(&shh[grp * 256 + ((bits >> shift) & 255u)], 1u);
            } else {
              if (bits > thrA[grp]) { gs[grp] += loss; gc[grp] += 1u; }
            }
          }
        }
      }
    }
  }

  if (mode == 0) {
    __syncthreads();
    for (int i = tid; i < 768; i += blockDim.x) {
      unsigned v = shh[i];
      if (v) atomicAdd(&hist[i], v);   // integer -> deterministic
    }
  } else {
    // deterministic float reduction: wave shuffle -> LDS -> fixed order
    #pragma unroll
    for (int off = 16; off > 0; off >>= 1) {
      #pragma unroll
      for (int gidx = 0; gidx < 3; ++gidx) {
        gs[gidx] += __shfl_down(gs[gidx], off, 32);
        gc[gidx] += __shfl_down(gc[gidx], off, 32);
      }
    }
    if (lane == 0) {
      #pragma unroll
      for (int gidx = 0; gidx < 3; ++gidx) {
        wsum[wid][gidx] = gs[gidx];
        wcnt[wid][gidx] = gc[gidx];
      }
    }
    __syncthreads();
    if (tid == 0) {
      for (int gidx = 0; gidx < 3; ++gidx) {
        float    s = 0.f;
        unsigned c = 0u;
        for (int w = 0; w < WPB; ++w) { s += wsum[w][gidx]; c += wcnt[w][gidx]; }
        partial[(size_t)blockIdx.x * 3 + gidx] = s;
        if (c) atomicAdd(&state[15 + gidx], c);   // integer -> deterministic
      }
    }
  }
}

// ---------------------------------------------------------------- radix scan
// state: [0..2]=count [3..5]=keep [6..8]=kneed [9..11]=prefix
//        [12..14]=threshold bits [15..17]=count_above
__global__ void hp_scan_kernel(unsigned* hist, unsigned* state, int pass) {
  if (threadIdx.x != 0) return;
  const int shift = 24 - 8 * pass;
  const double frac[3] = {0.5, 0.5, 0.1};   // F, S, O
  for (int g = 0; g < 3; ++g) {
    unsigned* h = hist + g * 256;
    unsigned kneed;
    if (pass == 0) {
      unsigned long long tot = 0ull;
      for (int b = 0; b < 256; ++b) tot += h[b];
      unsigned count = (unsigned)tot;
      state[g] = count;
      unsigned keep = 0u;
      if (count > 0u) {
        keep = (unsigned)ceil((double)count * frac[g]);
        if (keep < 32u) keep = 32u;
        if (keep > count) keep = count;
      }
      state[3 + g] = keep;
      kneed = keep;
    } else {
      kneed = state[6 + g];
    }
    if (state[g] > 0u) {
      unsigned cum = 0u;
      int digit = 0;
      for (int b = 255; b >= 0; --b) {
        cum += h[b];
        if (cum >= kneed) { digit = b; break; }
      }
      unsigned above = cum - h[digit];
      state[6 + g]  = kneed - above;
      state[9 + g] |= ((unsigned)digit) << shift;
    }
    for (int b = 0; b < 256; ++b) h[b] = 0u;   // reset for next pass
  }
  if (pass == 3) {
    for (int g = 0; g < 3; ++g) {
      state[12 + g] = state[9 + g];   // exact k-th largest value bits
      state[15 + g] = 0u;             // reset count_above for SUM pass
    }
  }
}

// ---------------------------------------------------------------- finalize
__global__ void hp_final_kernel(const float* __restrict__ partial, int nblk,
                                const unsigned* __restrict__ state,
                                float* __restrict__ out) {
  __shared__ float sh0[256], sh1[256], sh2[256];
  const int tid = threadIdx.x;
  float s0 = 0.f, s1 = 0.f, s2 = 0.f;
  for (int b = tid; b < nblk; b += 256) {
    s0 += partial[(size_t)b * 3 + 0];
    s1 += partial[(size_t)b * 3 + 1];
    s2 += partial[(size_t)b * 3 + 2];
  }
  sh0[tid] = s0; sh1[tid] = s1; sh2[tid] = s2;
  __syncthreads();
  for (int off = 128; off > 0; off >>= 1) {
    if (tid < off) {
      sh0[tid] += sh0[tid + off];
      sh1[tid] += sh1[tid + off];
      sh2[tid] += sh2[tid + off];
    }
    __syncthreads();
  }
  if (tid == 0) {
    float sums[3] = {sh0[0], sh1[0], sh2[0]};
    const float W[3] = {2.0f, 1.5f, 0.25f};   // F_W, S_W, O_W
    float num = 0.f, den = 0.f;
    for (int g = 0; g < 3; ++g) {
      unsigned count = state[g];
      if (count > 0u) {
        unsigned keep = state[3 + g];
        unsigned cab  = state[15 + g];
        float thr = __uint_as_float(state[12 + g]);
        float topsum = sums[g] + (float)(keep - cab) * thr;  // exact ties
        float mean = topsum / (float)keep;
        num += W[g] * mean;
        den += W[g];
      }
    }
    out[0] = (den > 0.f) ? (num / fmaxf(den, EPSF)) : 0.f;
  }
}

// ---------------------------------------------------------------- launch
extern "C" void kernel_launch(void* const* d_in, const int* in_sizes, int n_in,
                              void* d_out, int out_size, void* d_ws, size_t ws_size,
                              hipStream_t stream) {
  (void)in_sizes; (void)n_in; (void)out_size; (void)ws_size;
  const float*     E   = (const float*)d_in[0];
  const long long* cid = (const long long*)d_in[1];
  const float*     W1  = (const float*)d_in[2];
  const float*     b1  = (const float*)d_in[3];
  const float*     W2  = (const float*)d_in[4];
  const float*     b2  = (const float*)d_in[5];
  float*           out = (float*)d_out;

  char* ws = (char*)d_ws;
  float*    X       = (float*)ws;                      // 4096*64*4 = 1048576
  float*    sqv     = (float*)(ws + 1048576);          // 16384
  unsigned* codes   = (unsigned*)(ws + 1064960);       // 16384
  unsigned* hist    = (unsigned*)(ws + 1081344);       // 3*256*4 = 3072
  unsigned* state   = (unsigned*)(ws + 1084416);       // 32*4 = 128
  float*    partial = (float*)(ws + 1084544);          // NBLK2*3*4 = 12384

  hp_init_kernel<<<1, 256, 0, stream>>>(hist, state);
  hp_proj_kernel<<<NT, 128, 0, stream>>>(E, cid, W1, b1, W2, b2, X, sqv, codes);
  for (int pass = 0; pass < 4; ++pass) {
    hp_pair_kernel<<<NBLK2, WPB * 32, 0, stream>>>(X, sqv, codes, hist, state,
                                                   partial, 0, pass);
    hp_scan_kernel<<<1, 32, 0, stream>>>(hist, state, pass);
  }
  hp_pair_kernel<<<NBLK2, WPB * 32, 0, stream>>>(X, sqv, codes, hist, state,
                                                 partial, 1, 0);
  hp_final_kernel<<<1, 256, 0, stream>>>(partial, NBLK2, state, out);
}